// SPDNet3BiRe_4432406250096
// MI455X (gfx1250) — compile-verified
//
#include <hip/hip_runtime.h>
#include <math.h>
#include <stdint.h>

// SPDNet (BiMap -> ReEig) x3 -> LogEig -> vech -> FC, for MI455X (gfx1250).
// - All GEMM-shaped math uses V_WMMA_F32_16X16X4_F32 (fp32 matrix core): the
//   Jacobi eigensolver downstream needs full fp32 precision.
// - X tiles are staged global->LDS with CDNA5 async copies
//   (GLOBAL_LOAD_ASYNC_TO_LDS_B128, ASYNCcnt) so the WMMA MAC loops feed
//   exclusively from LDS at B128 granularity.
// - Eigendecomposition: parallel two-sided Jacobi in LDS (chess-tournament
//   ordering, D/2 disjoint rotations/round), spectral map, WMMA reconstruct.

typedef __attribute__((ext_vector_type(2))) float v2f;
typedef __attribute__((ext_vector_type(8))) float v8f;

#define EPS_REEIG 0.001f
#define JACOBI_SWEEPS 8

__device__ __forceinline__ v8f wmma_f32_16x16x4(v2f a, v2f b, v8f c) {
  // 8 args: (neg_a, A, neg_b, B, c_mod, C, reuse_a, reuse_b)
  return __builtin_amdgcn_wmma_f32_16x16x4_f32(false, a, false, b, (short)0, c,
                                               false, false);
}

// Async copy of 16 bytes global -> LDS (GV mode, tracked with ASYNCcnt).
__device__ __forceinline__ void async_copy_b128(const float* g, float* lds) {
  asm volatile("global_load_async_to_lds_b128 %0, %1, off"
               :
               : "v"((unsigned)(uintptr_t)lds), "v"(g)
               : "memory");
}
__device__ __forceinline__ void async_wait0() {
  asm volatile("s_wait_asynccnt 0x0" ::: "memory");
}

// ---------------------------------------------------------------------------
// BiMap: H[n] = W @ X[n] @ W^T     (W: NO x NI, X: NI x NI, H: NO x NO)
// One workgroup per matrix. W, X[n] and the intermediate T = W@X live in
// dynamic LDS; X[n] arrives via async B128 copies overlapped with the W load.
// ---------------------------------------------------------------------------
template <int NO, int NI>
__global__ void k_bimap(const float* __restrict__ W, const float* __restrict__ X,
                        float* __restrict__ H) {
  extern __shared__ float smem[];
  float* sW = smem;                // NO*NI
  float* sT = smem + NO * NI;      // NO*NI
  float* sX = smem + 2 * NO * NI;  // NI*NI

  const size_t n = blockIdx.x;
  const float* Xn = X + n * (size_t)(NI * NI);
  float* Hn = H + n * (size_t)(NO * NO);
  const int tid = threadIdx.x;

  // Kick off async X[n] -> LDS (B128 per lane), overlap with W load.
  for (int i = tid * 4; i < NI * NI; i += blockDim.x * 4)
    async_copy_b128(Xn + i, sX + i);
  for (int i = tid; i < NO * NI; i += blockDim.x) sW[i] = W[i];
  async_wait0();
  __syncthreads();

  const int wave = tid >> 5;
  const int nwav = blockDim.x >> 5;
  const int lane = tid & 31;
  const int half = lane >> 4;  // 0: K+0/K+1, 1: K+2/K+3 (f32 A/B frag layout)
  const int l16 = lane & 15;
  const int TM = NO / 16, TN = NI / 16;

  // Stage A: T = W @ X  (NO x NI), all operands in LDS.
  for (int t = wave; t < TM * TN; t += nwav) {
    const int tm = t / TN, tn = t % TN;
    v8f acc = {};
    const float* aRow = sW + (tm * 16 + l16) * NI;
    const float* bCol = sX + tn * 16 + l16;
#pragma unroll 4
    for (int k = 0; k < NI; k += 4) {
      v2f a, b;
      a.x = aRow[k + 2 * half];
      a.y = aRow[k + 2 * half + 1];
      b.x = bCol[(k + 2 * half) * NI];
      b.y = bCol[(k + 2 * half + 1) * NI];
      acc = wmma_f32_16x16x4(a, b, acc);
    }
#pragma unroll
    for (int r = 0; r < 8; ++r)
      sT[(tm * 16 + r + 8 * half) * NI + tn * 16 + l16] = acc[r];
  }
  __syncthreads();

  // Stage B: H = T @ W^T  (NO x NO); B[k][j] = W[j][k] read from LDS.
  for (int t = wave; t < TM * TM; t += nwav) {
    const int tm = t / TM, tn = t % TM;
    v8f acc = {};
    const float* aRow = sT + (tm * 16 + l16) * NI;
    const float* bRow = sW + (tn * 16 + l16) * NI;
#pragma unroll 4
    for (int k = 0; k < NI; k += 4) {
      v2f a, b;
      a.x = aRow[k + 2 * half];
      a.y = aRow[k + 2 * half + 1];
      b.x = bRow[k + 2 * half];
      b.y = bRow[k + 2 * half + 1];
      acc = wmma_f32_16x16x4(a, b, acc);
    }
#pragma unroll
    for (int r = 0; r < 8; ++r)
      Hn[(tm * 16 + r + 8 * half) * NO + tn * 16 + l16] = acc[r];
  }
}

// ---------------------------------------------------------------------------
// Jacobi eigensolver + spectral map + WMMA reconstruction, in place.
// take_log==0:  M := V diag(max(w,eps)) V^T          (ReEig)
// take_log==1:  M := V diag(log(max(w,eps))) V^T     (ReEig+LogEig fused:
//               reeig and logeig share eigenvectors, so one eigh suffices)
// ---------------------------------------------------------------------------
template <int D>
__global__ void k_jacobi_reeig(float* __restrict__ M, float eps, int take_log) {
  extern __shared__ float smem[];
  float* sA = smem;                  // D*D   working matrix
  float* sV = sA + D * D;            // D*D   eigenvectors
  float* sE = sV + D * D;            // D     mapped eigenvalues
  float* sC = sE + D;                // D/2   cos per pair
  float* sS = sC + D / 2;            // D/2   sin per pair
  int* sTop = (int*)(sS + D / 2);    // D/2
  int* sBot = sTop + D / 2;          // D/2

  const int tid = threadIdx.x;
  const int bs = blockDim.x;
  const int m = D / 2;
  float* Mn = M + (size_t)blockIdx.x * (D * D);

  // Async stage of the matrix into LDS; V := I meanwhile.
  for (int i = tid * 4; i < D * D; i += bs * 4) async_copy_b128(Mn + i, sA + i);
  for (int i = tid; i < D * D; i += bs)
    sV[i] = ((i / D) == (i % D)) ? 1.0f : 0.0f;
  if (tid < m) {
    sTop[tid] = 2 * tid;
    sBot[tid] = 2 * tid + 1;
  }
  async_wait0();
  __syncthreads();

  for (int sweep = 0; sweep < JACOBI_SWEEPS; ++sweep) {
    for (int round = 0; round < D - 1; ++round) {
      // 1) rotation angles for the m disjoint pairs
      if (tid < m) {
        int p = sTop[tid], q = sBot[tid];
        if (p > q) { int t = p; p = q; q = t; }
        float app = sA[p * D + p], aqq = sA[q * D + q], apq = sA[p * D + q];
        float c = 1.0f, s = 0.0f;
        if (fabsf(apq) > 1e-12f) {
          float tau = (aqq - app) / (2.0f * apq);
          float tt = ((tau >= 0.0f) ? 1.0f : -1.0f) /
                     (fabsf(tau) + sqrtf(1.0f + tau * tau));
          c = rsqrtf(1.0f + tt * tt);
          s = tt * c;
        }
        sC[tid] = c;
        sS[tid] = s;
      }
      __syncthreads();
      // 2) row rotations (pairs touch disjoint rows)
      for (int it = tid; it < m * D; it += bs) {
        const int pr = it / D, k = it % D;
        int p = sTop[pr], q = sBot[pr];
        if (p > q) { int t = p; p = q; q = t; }
        const float c = sC[pr], s = sS[pr];
        const float ap = sA[p * D + k], aq = sA[q * D + k];
        sA[p * D + k] = c * ap - s * aq;
        sA[q * D + k] = s * ap + c * aq;
      }
      __syncthreads();
      // 3) column rotations on A, and accumulate V
      for (int it = tid; it < m * D; it += bs) {
        const int pr = it / D, k = it % D;
        int p = sTop[pr], q = sBot[pr];
        if (p > q) { int t = p; p = q; q = t; }
        const float c = sC[pr], s = sS[pr];
        const float ap = sA[k * D + p], aq = sA[k * D + q];
        sA[k * D + p] = c * ap - s * aq;
        sA[k * D + q] = s * ap + c * aq;
        const float vp = sV[k * D + p], vq = sV[k * D + q];
        sV[k * D + p] = c * vp - s * vq;
        sV[k * D + q] = s * vp + c * vq;
      }
      __syncthreads();
      // 4) chess-tournament permutation (top[0] fixed)
      if (tid == 0) {
        const int oldTopLast = sTop[m - 1];
        const int oldBot0 = sBot[0];
        for (int i = m - 1; i >= 2; --i) sTop[i] = sTop[i - 1];
        if (m > 1) sTop[1] = oldBot0;
        for (int i = 0; i < m - 1; ++i) sBot[i] = sBot[i + 1];
        sBot[m - 1] = oldTopLast;
      }
      __syncthreads();
    }
  }

  // Spectral map on the (near-)diagonal
  for (int i = tid; i < D; i += bs) {
    const float w = fmaxf(sA[i * D + i], eps);
    sE[i] = take_log ? logf(w) : w;
  }
  __syncthreads();
  // sA := V * diag(f(w))
  for (int i = tid; i < D * D; i += bs) sA[i] = sV[i] * sE[i % D];
  __syncthreads();

  // Mn := sA @ V^T via fp32 WMMA (wave-uniform control flow => EXEC all 1s)
  const int wave = tid >> 5, nwav = bs >> 5;
  const int lane = tid & 31, half = lane >> 4, l16 = lane & 15;
  const int T16 = D / 16;
  for (int t = wave; t < T16 * T16; t += nwav) {
    const int tm = t / T16, tn = t % T16;
    v8f acc = {};
    const float* aRow = sA + (tm * 16 + l16) * D;
    const float* bRow = sV + (tn * 16 + l16) * D;  // B[k][j] = V[j][k]
    for (int k = 0; k < D; k += 4) {
      v2f a, b;
      a.x = aRow[k + 2 * half];
      a.y = aRow[k + 2 * half + 1];
      b.x = bRow[k + 2 * half];
      b.y = bRow[k + 2 * half + 1];
      acc = wmma_f32_16x16x4(a, b, acc);
    }
#pragma unroll
    for (int r = 0; r < 8; ++r)
      Mn[(tm * 16 + r + 8 * half) * D + tn * 16 + l16] = acc[r];
  }
}

// ---------------------------------------------------------------------------
// vech (upper-tri, off-diag * sqrt2, row-major triu order) + FC [136 -> 4]
// ---------------------------------------------------------------------------
__global__ void k_vechfc(const float* __restrict__ G, const float* __restrict__ fcw,
                         const float* __restrict__ fcb, float* __restrict__ out,
                         int N) {
  const int idx = blockIdx.x * blockDim.x + threadIdx.x;
  if (idx >= N * 4) return;
  const int n = idx >> 2, c = idx & 3;
  const float* g = G + (size_t)n * 256;
  const float* wr = fcw + c * 136;
  const float r2 = 1.41421356237309515f;
  float acc = fcb[c];
  int t = 0;
  for (int i = 0; i < 16; ++i)
    for (int j = i; j < 16; ++j) {
      const float sc = (i == j) ? 1.0f : r2;
      acc += wr[t++] * g[i * 16 + j] * sc;
    }
  out[idx] = acc;
}

// ---------------------------------------------------------------------------
extern "C" void kernel_launch(void* const* d_in, const int* in_sizes, int n_in,
                              void* d_out, int out_size, void* d_ws, size_t ws_size,
                              hipStream_t stream) {
  (void)n_in; (void)out_size; (void)ws_size;
  const float* x = (const float*)d_in[0];    // [N,128,128]
  const float* W1 = (const float*)d_in[1];   // [64,128]
  const float* W2 = (const float*)d_in[2];   // [32,64]
  const float* W3 = (const float*)d_in[3];   // [16,32]
  const float* fcw = (const float*)d_in[4];  // [4,136]
  const float* fcb = (const float*)d_in[5];  // [4]
  float* out = (float*)d_out;                // [N,4]
  const int N = in_sizes[0] / (128 * 128);

  float* buf1 = (float*)d_ws;                // N*64*64  (128 MB)
  float* buf2 = buf1 + (size_t)N * 64 * 64;  // N*32*32  ( 32 MB)
  float* buf3 = buf2 + (size_t)N * 32 * 32;  // N*16*16  (  8 MB)

  size_t sm;
  sm = (size_t)(2 * 64 * 128 + 128 * 128) * sizeof(float);  // 128 KB
  k_bimap<64, 128><<<N, 256, sm, stream>>>(W1, x, buf1);
  sm = (size_t)(2 * 64 * 64 + 3 * 64) * sizeof(float);
  k_jacobi_reeig<64><<<N, 256, sm, stream>>>(buf1, EPS_REEIG, 0);

  sm = (size_t)(2 * 32 * 64 + 64 * 64) * sizeof(float);     // 32 KB
  k_bimap<32, 64><<<N, 256, sm, stream>>>(W2, buf1, buf2);
  sm = (size_t)(2 * 32 * 32 + 3 * 32) * sizeof(float);
  k_jacobi_reeig<32><<<N, 128, sm, stream>>>(buf2, EPS_REEIG, 0);

  sm = (size_t)(2 * 16 * 32 + 32 * 32) * sizeof(float);     // 8 KB
  k_bimap<16, 32><<<N, 64, sm, stream>>>(W3, buf2, buf3);
  // Fused ReEig+LogEig (shared eigenvectors): log(max(w, eps))
  sm = (size_t)(2 * 16 * 16 + 3 * 16) * sizeof(float);
  k_jacobi_reeig<16><<<N, 64, sm, stream>>>(buf3, EPS_REEIG, 1);

  const int threads = 256;
  const int blocks = (N * 4 + threads - 1) / threads;
  k_vechfc<<<blocks, threads, 0, stream>>>(buf3, fcw, fcb, out, N);
}